// GCEGNN_46351287058916
// MI455X (gfx1250) — compile-verified
//
#include <hip/hip_runtime.h>
#include <hip/hip_bf16.h>

typedef __attribute__((ext_vector_type(2))) float v2f;
typedef __attribute__((ext_vector_type(8))) float v8f;

#define SL 50          // max session length
#define DM 128         // d_model
#define NB 256         // batch
#define V1 500001      // V+1

// ---------------- Kernel A LDS layout (word offsets) ----------------
// big buffers are [64][132] floats (M padded to 64, pitch 132 -> 4-bank step, conflict-free)
constexpr int PIT   = 132;
constexpr int BIGF  = 64 * PIT;          // 8448 words per big buffer
constexpr int APIT  = 52;                // adjacency pitch
constexpr int OFF_HH    = 0;
constexpr int OFF_B1    = 1 * BIGF;
constexpr int OFF_B2    = 2 * BIGF;
constexpr int OFF_B3    = 3 * BIGF;
constexpr int OFF_B4    = 4 * BIGF;
constexpr int OFF_AIN   = 5 * BIGF;
constexpr int OFF_AOUT  = OFF_AIN  + SL * APIT;
constexpr int OFF_SI    = OFF_AOUT + SL * APIT;   // int[64]
constexpr int OFF_INV   = OFF_SI   + 64;          // int[64]
constexpr int OFF_UNIQ  = OFF_INV  + 64;          // int[64]
constexpr int OFF_ATT   = OFF_UNIQ + 64;          // float[64]
constexpr int OFF_ALPHA = OFF_ATT  + 64;          // float[64]
constexpr int OFF_SG    = OFF_ALPHA+ 64;          // float[128]
constexpr int OFF_LAST  = OFF_SG   + 128;         // float[128]
constexpr int OFF_SCAL  = OFF_LAST + 128;         // int[8]
constexpr int TOTAL_A_WORDS = OFF_SCAL + 8;
constexpr int LDSA_BYTES = TOTAL_A_WORDS * 4;     // ~192 KB (<= 320 KB WGP LDS)

constexpr int LDSB_BYTES = (128 * 129 + 256 * 129) * 4;  // ~194 KB

__device__ __forceinline__ float sigmoidf_(float x) { return 1.0f / (1.0f + expf(-x)); }

// C_lds[64][PIT] = A1_lds[64][PIT] @ W1[128x128]  (+ A2_lds @ W2 if A2 != nullptr)
// 4 waves per block; each wave does 8 of the 32 (mt,nt) 16x16 tiles, K=128 in 32 WMMA steps.
__device__ __forceinline__ void wave_gemm(const float* __restrict__ A1,
                                          const float* __restrict__ W1,
                                          const float* __restrict__ A2,
                                          const float* __restrict__ W2,
                                          float* __restrict__ C,
                                          int wave, int lane) {
  const int hi = lane >> 4;   // 0/1: which K pair / which M half for C
  const int lo = lane & 15;
  for (int t = wave; t < 32; t += 4) {
    const int mt = t >> 3;    // 0..3
    const int nt = t & 7;     // 0..7
    const int ncol = nt * 16 + lo;
    v8f c = {0.f,0.f,0.f,0.f,0.f,0.f,0.f,0.f};
    {
      const float* Arow = A1 + (mt * 16 + lo) * PIT;
      for (int kk = 0; kk < 32; ++kk) {
        const int k0 = kk * 4 + hi * 2;
        v2f a = {Arow[k0], Arow[k0 + 1]};
        v2f b = {W1[k0 * DM + ncol], W1[(k0 + 1) * DM + ncol]};
        c = __builtin_amdgcn_wmma_f32_16x16x4_f32(false, a, false, b, (short)0, c, false, false);
      }
    }
    if (A2 != nullptr) {
      const float* Arow = A2 + (mt * 16 + lo) * PIT;
      for (int kk = 0; kk < 32; ++kk) {
        const int k0 = kk * 4 + hi * 2;
        v2f a = {Arow[k0], Arow[k0 + 1]};
        v2f b = {W2[k0 * DM + ncol], W2[(k0 + 1) * DM + ncol]};
        c = __builtin_amdgcn_wmma_f32_16x16x4_f32(false, a, false, b, (short)0, c, false, false);
      }
    }
    // C layout: VGPR r -> row (mt*16 + 8*hi + r), col ncol
    float* Crow = C + (mt * 16 + hi * 8) * PIT + nt * 16 + lo;
#pragma unroll
    for (int r = 0; r < 8; ++r) Crow[r * PIT] = c[r];
  }
}

// ---------------- Kernel A: per-session graph + GNN -> reps[256][128] ----------------
__global__ __launch_bounds__(128)
void gce_session_kernel(const int* __restrict__ seq, const float* __restrict__ emb,
                        const float* __restrict__ W_in, const float* __restrict__ W_out,
                        const float* __restrict__ Wz, const float* __restrict__ bz,
                        const float* __restrict__ Uz,
                        const float* __restrict__ Wr, const float* __restrict__ br,
                        const float* __restrict__ Ur,
                        const float* __restrict__ Wh, const float* __restrict__ bh,
                        const float* __restrict__ Uh,
                        const float* __restrict__ Wg, const float* __restrict__ bg,
                        const float* __restrict__ Wgate, const float* __restrict__ bgate,
                        const float* __restrict__ Wproj, const float* __restrict__ bproj,
                        float* __restrict__ reps) {
  extern __shared__ float sm[];
  float* Hh   = sm + OFF_HH;
  float* B1   = sm + OFF_B1;
  float* B2   = sm + OFF_B2;
  float* B3   = sm + OFF_B3;
  float* B4   = sm + OFF_B4;
  float* Ain  = sm + OFF_AIN;
  float* Aout = sm + OFF_AOUT;
  int*   sI   = (int*)(sm + OFF_SI);
  int*   inv  = (int*)(sm + OFF_INV);
  int*   uq   = (int*)(sm + OFF_UNIQ);
  float* att  = sm + OFF_ATT;
  float* alp  = sm + OFF_ALPHA;
  float* sg   = sm + OFF_SG;
  float* lst  = sm + OFF_LAST;
  int*   scal = (int*)(sm + OFF_SCAL);

  const int tid  = threadIdx.x;       // 0..127
  const int wave = tid >> 5;
  const int lane = tid & 31;
  const int bss  = blockIdx.x;

  if (tid < SL) sI[tid] = seq[bss * SL + tid];
  if (tid < SL) uq[tid] = 0;
  __syncthreads();

  // ---- unique (sorted) / inverse indices ----
  if (tid < SL) {
    const int v = sI[tid];
    int rank = 0;
    if (v > 0) {
      for (int j = 0; j < SL; ++j) {
        const int u = sI[j];
        if (u > 0 && u < v) {
          bool first = true;
          for (int k = 0; k < j; ++k) if (sI[k] == u) { first = false; break; }
          rank += first ? 1 : 0;
        }
      }
    }
    inv[tid] = (v > 0) ? rank : 0;
  }
  if (tid == 0) {
    int nv = 0, nn = 0;
    for (int i = 0; i < SL; ++i) {
      const int v = sI[i];
      if (v > 0) {
        ++nv;
        bool first = true;
        for (int j = 0; j < i; ++j) if (sI[j] == v) { first = false; break; }
        nn += first ? 1 : 0;
      }
    }
    scal[0] = nv; scal[1] = nn;
  }
  __syncthreads();
  if (tid < SL) {
    const int v = sI[tid];
    if (v > 0) {
      bool first = true;
      for (int j = 0; j < tid; ++j) if (sI[j] == v) { first = false; break; }
      if (first) uq[inv[tid]] = v;
    }
  }
  __syncthreads();
  const int n_valid = scal[0];
  const int n_nodes = scal[1];

  // ---- adjacency (binary, row-normalized) ----
  for (int i = tid; i < SL * APIT; i += 128) { Ain[i] = 0.f; Aout[i] = 0.f; }
  __syncthreads();
  if (tid < SL - 1 && n_nodes > 1) {
    if (sI[tid] > 0 && sI[tid + 1] > 0) {
      const int s = inv[tid], d = inv[tid + 1];
      Aout[s * APIT + d] = 1.0f;
      Ain[d * APIT + s]  = 1.0f;
    }
  }
  __syncthreads();
  if (tid < SL) {
    float si = 0.f, so = 0.f;
    for (int j = 0; j < SL; ++j) { si += Ain[tid * APIT + j]; so += Aout[tid * APIT + j]; }
    const float ri = 1.0f / (si + 1e-8f), ro = 1.0f / (so + 1e-8f);
    for (int j = 0; j < SL; ++j) { Ain[tid * APIT + j] *= ri; Aout[tid * APIT + j] *= ro; }
  }

  // ---- h = emb[uniq], pad rows 50..63 with zero ----
  for (int idx = tid; idx < 64 * DM; idx += 128) {
    const int n = idx >> 7, d = idx & 127;
    float v = 0.f;
    if (n < SL) v = emb[(long)uq[n] * DM + d];   // uq==0 -> zero embedding row
    Hh[n * PIT + d] = v;
  }
  __syncthreads();

  // ---- t1 = h@W_in -> B1 ; t2 = h@W_out -> B2 ----
  wave_gemm(Hh, W_in,  nullptr, nullptr, B1, wave, lane);
  wave_gemm(Hh, W_out, nullptr, nullptr, B2, wave, lane);
  __syncthreads();

  // ---- m = A_in@t1 + A_out@t2 -> B3 (rows >=50 zero) ----
  for (int idx = tid; idx < 64 * DM; idx += 128) {
    const int n = idx >> 7, d = idx & 127;
    float acc = 0.f;
    if (n < SL) {
      for (int j = 0; j < SL; ++j)
        acc += Ain[n * APIT + j] * B1[j * PIT + d] + Aout[n * APIT + j] * B2[j * PIT + d];
    }
    B3[n * PIT + d] = acc;
  }
  __syncthreads();

  // ---- z = sigmoid(m@Wz + h@Uz + bz) -> B4 ----
  wave_gemm(B3, Wz, Hh, Uz, B4, wave, lane);
  __syncthreads();
  for (int idx = tid; idx < SL * DM; idx += 128) {
    const int n = idx >> 7, d = idx & 127;
    B4[n * PIT + d] = sigmoidf_(B4[n * PIT + d] + bz[d]);
  }
  __syncthreads();

  // ---- r -> rh = r*h in B1 ----
  wave_gemm(B3, Wr, Hh, Ur, B1, wave, lane);
  __syncthreads();
  for (int idx = tid; idx < 64 * DM; idx += 128) {
    const int n = idx >> 7, d = idx & 127;
    const float rv = (n < SL) ? sigmoidf_(B1[n * PIT + d] + br[d]) : 0.f;
    B1[n * PIT + d] = rv * Hh[n * PIT + d];
  }
  __syncthreads();

  // ---- ht = tanh(m@Wh + rh@Uh + bh); h_local = (1-z)h + z*ht -> B2 ----
  wave_gemm(B3, Wh, B1, Uh, B2, wave, lane);
  __syncthreads();
  for (int idx = tid; idx < SL * DM; idx += 128) {
    const int n = idx >> 7, d = idx & 127;
    const float ht = tanhf(B2[n * PIT + d] + bh[d]);
    const float z  = B4[n * PIT + d];
    B2[n * PIT + d] = (1.0f - z) * Hh[n * PIT + d] + z * ht;
  }
  __syncthreads();

  // ---- global graph: tg = h@Wg -> B1 ; h_global -> B4 ----
  wave_gemm(Hh, Wg, nullptr, nullptr, B1, wave, lane);
  __syncthreads();
  {
    const int d = tid;   // 128 threads, one column each
    float cs = 0.f;
    for (int j = 0; j < n_nodes; ++j) cs += B1[j * PIT + d];
    sg[d] = cs;          // column sums of tg over valid nodes
  }
  __syncthreads();
  for (int idx = tid; idx < SL * DM; idx += 128) {
    const int n = idx >> 7, d = idx & 127;
    float hg;
    if (n_nodes > 1) {
      hg = (n < n_nodes)
         ? (sg[d] + (float)(n_nodes - 1) * bg[d] - B1[n * PIT + d]) / ((float)(n_nodes - 1) + 1e-8f)
         : 0.f;
    } else {
      hg = Hh[n * PIT + d];
    }
    B4[n * PIT + d] = hg;
  }
  __syncthreads();

  // ---- gather to sequence positions: hl -> B1, hg -> Hh (Hh and tg are dead now) ----
  for (int idx = tid; idx < 64 * DM; idx += 128) {
    const int i = idx >> 7, d = idx & 127;
    float hlv = 0.f, hgv = 0.f;
    if (i < SL) {
      const int node = inv[i];
      hlv = B2[node * PIT + d];
      hgv = B4[node * PIT + d];
    }
    B1[i * PIT + d] = hlv;   // overwrites tg (dead)
    Hh[i * PIT + d] = hgv;   // overwrites h (dead)
  }
  __syncthreads();

  // ---- gate pre-activation via WMMA: gpre = hl@Wgate[:128] + hg@Wgate[128:] -> B3 ----
  wave_gemm(B1, Wgate, Hh, Wgate + DM * DM, B3, wave, lane);
  __syncthreads();

  // ---- h_seq = g*hl + (1-g)*hg -> B3 ----
  for (int idx = tid; idx < SL * DM; idx += 128) {
    const int i = idx >> 7, d = idx & 127;
    const float g = sigmoidf_(B3[i * PIT + d] + bgate[d]);
    B3[i * PIT + d] = g * B1[i * PIT + d] + (1.0f - g) * Hh[i * PIT + d];
  }
  __syncthreads();

  // ---- attention pooling ----
  { const int li = n_valid - 1; lst[tid] = B3[li * PIT + tid]; }
  __syncthreads();
  if (tid < SL) {
    float a = -1e9f;
    if (sI[tid] > 0) {
      a = 0.f;
      for (int k = 0; k < DM; ++k) a += B3[tid * PIT + k] * lst[k];
    }
    att[tid] = a;
  }
  __syncthreads();
  if (tid == 0) {
    float mx = att[0];
    for (int i = 1; i < SL; ++i) mx = fmaxf(mx, att[i]);
    float s = 0.f;
    for (int i = 0; i < SL; ++i) { const float e = expf(att[i] - mx); alp[i] = e; s += e; }
    const float rs = 1.0f / s;
    for (int i = 0; i < SL; ++i) alp[i] *= rs;
  }
  __syncthreads();
  {
    const int d = tid;
    float acc = 0.f;
    for (int i = 0; i < SL; ++i) acc += alp[i] * B3[i * PIT + d];
    sg[d] = acc;
  }
  __syncthreads();
  {
    const int d = tid;
    float acc = bproj[d];
    for (int k = 0; k < DM; ++k) acc += sg[k] * Wproj[k * DM + d];
    for (int k = 0; k < DM; ++k) acc += lst[k] * Wproj[(DM + k) * DM + d];
    reps[bss * DM + d] = acc;
  }
}

// ---------------- Kernel B: logits = reps @ emb^T  (streaming WMMA GEMM) ----------------
__global__ __launch_bounds__(256)
void gce_logits_kernel(const float* __restrict__ reps, const float* __restrict__ emb,
                       float* __restrict__ out) {
  extern __shared__ float smB[];
  float* Bt = smB;                 // [128][129] emb strip (item-major, odd pitch)
  float* At = smB + 128 * 129;     // [256][129] reps

  const int tid  = threadIdx.x;    // 0..255 = 8 waves
  const int wave = tid >> 5;
  const int lane = tid & 31;
  const int hi   = lane >> 4;
  const int lo   = lane & 15;
  const long colBase = (long)blockIdx.x * 128;

  // stage emb strip (coalesced float4); clamp tail rows (stores are guarded later)
  for (int idx = tid; idx < 128 * 32; idx += 256) {
    const int li = idx >> 5;
    const int d4 = (idx & 31) << 2;
    long gi = colBase + li; if (gi > (long)(V1 - 1)) gi = V1 - 1;
    const float4 v = *(const float4*)(emb + gi * DM + d4);
    float* p = Bt + li * 129 + d4;
    p[0] = v.x; p[1] = v.y; p[2] = v.z; p[3] = v.w;
  }
  // stage all reps (L2-resident, 128 KB)
  for (int idx = tid; idx < 256 * 32; idx += 256) {
    const int r  = idx >> 5;
    const int d4 = (idx & 31) << 2;
    const float4 v = *(const float4*)(reps + r * DM + d4);
    float* p = At + r * 129 + d4;
    p[0] = v.x; p[1] = v.y; p[2] = v.z; p[3] = v.w;
  }
  __syncthreads();

  // preload this wave's entire B fragment strip (16 cols x K=128) into registers:
  // reused across all 16 M tiles -> halves LDS traffic in the compute phase.
  const float* Brow = Bt + (wave * 16 + lo) * 129;
  v2f bf[32];
#pragma unroll
  for (int kk = 0; kk < 32; ++kk) {
    const int k0 = kk * 4 + hi * 2;
    bf[kk][0] = Brow[k0];
    bf[kk][1] = Brow[k0 + 1];
  }

  const long ncol = colBase + wave * 16 + lo;

  for (int mt = 0; mt < 16; ++mt) {
    v8f c = {0.f,0.f,0.f,0.f,0.f,0.f,0.f,0.f};
    const float* Arow = At + (mt * 16 + lo) * 129;
#pragma unroll 8
    for (int kk = 0; kk < 32; ++kk) {
      const int k0 = kk * 4 + hi * 2;
      v2f a = {Arow[k0], Arow[k0 + 1]};
      c = __builtin_amdgcn_wmma_f32_16x16x4_f32(false, a, false, bf[kk], (short)0, c, false, false);
    }
    if (ncol < (long)V1) {
      float* orow = out + (long)(mt * 16 + hi * 8) * V1 + ncol;
#pragma unroll
      for (int r = 0; r < 8; ++r) orow[(long)r * V1] = c[r];
    }
  }
}

// ---------------- host launcher ----------------
extern "C" void kernel_launch(void* const* d_in, const int* in_sizes, int n_in,
                              void* d_out, int out_size, void* d_ws, size_t ws_size,
                              hipStream_t stream) {
  const int*   seq   = (const int*)  d_in[0];
  const float* emb   = (const float*)d_in[1];
  const float* W_in  = (const float*)d_in[2];
  const float* W_out = (const float*)d_in[3];
  const float* Wz    = (const float*)d_in[4];
  const float* bz    = (const float*)d_in[5];
  const float* Uz    = (const float*)d_in[6];
  const float* Wr    = (const float*)d_in[7];
  const float* br    = (const float*)d_in[8];
  const float* Ur    = (const float*)d_in[9];
  const float* Wh    = (const float*)d_in[10];
  const float* bh    = (const float*)d_in[11];
  const float* Uh    = (const float*)d_in[12];
  const float* Wg    = (const float*)d_in[13];
  const float* bg    = (const float*)d_in[14];
  const float* Wgate = (const float*)d_in[15];
  const float* bgate = (const float*)d_in[16];
  const float* Wproj = (const float*)d_in[17];
  const float* bproj = (const float*)d_in[18];

  float* reps = (float*)d_ws;   // 256*128 floats

  (void)in_sizes; (void)n_in; (void)out_size; (void)ws_size;

  hipFuncSetAttribute(reinterpret_cast<const void*>(gce_session_kernel),
                      hipFuncAttributeMaxDynamicSharedMemorySize, LDSA_BYTES);
  hipFuncSetAttribute(reinterpret_cast<const void*>(gce_logits_kernel),
                      hipFuncAttributeMaxDynamicSharedMemorySize, LDSB_BYTES);

  gce_session_kernel<<<NB, 128, LDSA_BYTES, stream>>>(
      seq, emb, W_in, W_out, Wz, bz, Uz, Wr, br, Ur, Wh, bh, Uh,
      Wg, bg, Wgate, bgate, Wproj, bproj, reps);

  const int nColBlocks = (V1 + 127) / 128;  // 3907
  gce_logits_kernel<<<nColBlocks, 256, LDSB_BYTES, stream>>>(reps, emb, (float*)d_out);
}